// StationaryHMM_42563125903453
// MI455X (gfx1250) — compile-verified
//
#include <hip/hip_runtime.h>

// ============================================================================
// StationaryHMM forward filter for MI455X (gfx1250, wave32).
//
// The filter is a linear recurrence over 16-vectors:
//     alpha_t = alpha_{t-1} * (P diag(g_t))        (up to normalization)
// Matrix products are associative -> 3-phase parallel scan with
// V_WMMA_F32_16X16X4_F32 (16x16x16 product = 4 chained K=4 WMMAs):
//   k_chunkprod : per-chunk products N_c (parallel over 710 chunks)
//   k_scan      : sequential scan of chunk matrices -> chunk start states
//   k_emit      : 16 chunks per wave as a 16x16 state matrix, emits outputs
// D-layout -> B-layout conversion per step uses ds_swizzle SWAPX16.
// Phase-1 rescale uses a cheap wave-uniform scale (column-0 sum via one
// SWAP16 + v_readfirstlane) instead of a 5-level reduction tree: the scale
// cancels in all later normalizations, so any uniform positive value works.
//
// d_out layout (float): ut[0 .. 16T), u_norm[16T .. 32T), ft[32T .. 33T)
// Workspace need: ~776 KB (P, P^T, params, chunk states, chunk matrices).
// ============================================================================

#define T_LEN   500000
#define KS      16
#define CHUNK   704
#define NCH     ((T_LEN + CHUNK - 1) / CHUNK)   // 711

#define OFF_UT  0
#define OFF_UN  (16 * T_LEN)
#define OFF_FT  (32 * T_LEN)

// workspace float offsets
#define WS_P      0                     // P row-major        [16][16]
#define WS_PT     256                   // P^T row-major      [16][16]
#define WS_MU     512                   // mu                 [16]
#define WS_IS     528                   // 1/sigma            [16]
#define WS_CF     544                   // inv_sqrt_2pi/sigma [16]
#define WS_STATE  576                   // chunk start states [NCH][16]
#define WS_N      (WS_STATE + NCH * 16) // chunk matrices N^T [NCH][16][16]

typedef float v2f __attribute__((ext_vector_type(2)));
typedef float v4f __attribute__((ext_vector_type(4)));
typedef float v8f __attribute__((ext_vector_type(8)));

// ds_swizzle group-of-32: offset = xor<<10 | or<<5 | and
#define SWZ(x, xorm) __builtin_bit_cast(float, \
    __builtin_amdgcn_ds_swizzle(__builtin_bit_cast(int, (x)), ((xorm) << 10) | 0x1f))
#define SWAP16(x) SWZ(x, 16)

__device__ __forceinline__ v8f wmma_k4(v2f a, v2f b, v8f c) {
  // D(16x16,f32) = A(16x4,f32) x B(4x16,f32) + C
  return __builtin_amdgcn_wmma_f32_16x16x4_f32(false, a, false, b, (short)0, c,
                                               false, false);
}

struct Asl { v2f a[4]; };  // 4 K-slices of a 16x16 A operand
struct Bsl { v2f b[4]; };  // 4 K-slices of a 16x16 B operand

// Convert a 16x16 matrix in WMMA C/D layout (VGPR v: lanes0-15 -> row v,
// lanes16-31 -> row v+8) into the four 4x16 B-layout slices
// (slice j VGPR0: lanes0-15 = row 4j, lanes16-31 = row 4j+2; VGPR1: 4j+1/4j+3).
__device__ __forceinline__ Bsl d_to_b(const v8f R, bool hi) {
  float w0 = SWAP16(R[0]), w1 = SWAP16(R[1]), w2 = SWAP16(R[2]), w3 = SWAP16(R[3]);
  float w4 = SWAP16(R[4]), w5 = SWAP16(R[5]), w6 = SWAP16(R[6]), w7 = SWAP16(R[7]);
  Bsl o;
  o.b[0].x = hi ? w2 : R[0];  o.b[0].y = hi ? w3 : R[1];   // rows 0..3
  o.b[1].x = hi ? w6 : R[4];  o.b[1].y = hi ? w7 : R[5];   // rows 4..7
  o.b[2].x = hi ? R[2] : w0;  o.b[2].y = hi ? R[3] : w1;   // rows 8..11
  o.b[3].x = hi ? R[6] : w4;  o.b[3].y = hi ? R[7] : w5;   // rows 12..15
  return o;
}

// Build A slices from a per-lane array col[16] holding row (lane&15) of A.
// A-layout slice j: VGPR0 = K=4j (lanes lo) / 4j+2 (lanes hi); VGPR1 = 4j+1/4j+3.
__device__ __forceinline__ Asl make_a(const float col[16], bool hi) {
  Asl s;
#pragma unroll
  for (int j = 0; j < 4; ++j) {
    s.a[j].x = hi ? col[4 * j + 2] : col[4 * j];
    s.a[j].y = hi ? col[4 * j + 3] : col[4 * j + 1];
  }
  return s;
}

__device__ __forceinline__ v8f mm_acc(const Asl& A, const Bsl& B) {
  v8f acc = {0.f, 0.f, 0.f, 0.f, 0.f, 0.f, 0.f, 0.f};
  acc = wmma_k4(A.a[0], B.b[0], acc);
  acc = wmma_k4(A.a[1], B.b[1], acc);
  acc = wmma_k4(A.a[2], B.b[2], acc);
  acc = wmma_k4(A.a[3], B.b[3], acc);
  return acc;
}

// ---------------------------------------------------------------------------
// Phase 0: softmax(P), params, stationary distribution (power iteration),
//          t=0 outputs and chunk-0 start state.  One wave.
// ---------------------------------------------------------------------------
__global__ void k_init(const float* __restrict__ y, const float* __restrict__ logits,
                       const float* __restrict__ mu, const float* __restrict__ logsig,
                       float* __restrict__ out, float* __restrict__ W) {
  __shared__ float Psh[KS * 17];
  __shared__ float vsh[KS];
  const int lane = threadIdx.x;
  const int m = lane & 15;

  float mu_l = 0.f, cf_l = 0.f, is_l = 0.f;
  if (lane < KS) {
    float r[KS];
#pragma unroll
    for (int k = 0; k < KS; ++k) r[k] = logits[lane * KS + k];
    float mx = r[0];
#pragma unroll
    for (int k = 1; k < KS; ++k) mx = fmaxf(mx, r[k]);
    float s = 0.f;
#pragma unroll
    for (int k = 0; k < KS; ++k) { r[k] = __expf(r[k] - mx); s += r[k]; }
    const float inv = 1.0f / s;
#pragma unroll
    for (int k = 0; k < KS; ++k) {
      const float p = r[k] * inv;
      W[WS_P + lane * KS + k] = p;     // P row-major
      W[WS_PT + k * KS + lane] = p;    // P^T row-major
      Psh[lane * 17 + k] = p;
    }
    mu_l = mu[lane];
    is_l = __expf(-logsig[lane]);
    cf_l = 0.3989422804014327f * is_l;
    W[WS_MU + lane] = mu_l;
    W[WS_IS + lane] = is_l;
    W[WS_CF + lane] = cf_l;
  }
  __syncthreads();

  float Pc[KS];  // column m of P, per lane
#pragma unroll
  for (int k = 0; k < KS; ++k) Pc[k] = Psh[k * 17 + m];

  // power iteration: pi <- normalize(pi P); converges to the unique stationary
  // distribution of the strictly-positive softmax matrix (== LS solution).
  float pi[KS];
#pragma unroll
  for (int k = 0; k < KS; ++k) pi[k] = 1.0f / 16.0f;
  for (int it = 0; it < 256; ++it) {
    float nn = 0.f;
#pragma unroll
    for (int k = 0; k < KS; ++k) nn += pi[k] * Pc[k];
    if (lane < KS) vsh[m] = nn;
    __syncthreads();
    float s = 0.f;
#pragma unroll
    for (int k = 0; k < KS; ++k) { pi[k] = vsh[k]; s += pi[k]; }
    const float inv = 1.0f / s;
#pragma unroll
    for (int k = 0; k < KS; ++k) pi[k] *= inv;
    __syncthreads();
  }

  // t = 0 special step
  const float y0 = y[0];
  if (lane < KS) {
    const float z = (y0 - mu_l) * is_l;
    vsh[m] = __expf(-0.5f * z * z) * cf_l;
  }
  __syncthreads();
  float ft0 = 0.f, g[KS];
#pragma unroll
  for (int k = 0; k < KS; ++k) { g[k] = vsh[k]; ft0 += pi[k] * g[k]; }
  const float invf = 1.0f / ft0;
  if (lane < KS) {
    const float utt0 = pi[m] * g[m] * invf;
    out[OFF_UT + m] = pi[m];
    out[OFF_UN + m] = utt0;
    W[WS_STATE + m] = utt0;
  }
  if (lane == 0) out[OFF_FT + 0] = ft0;
}

// ---------------------------------------------------------------------------
// Phase 1: chunk products.  Block c computes N_c^T = A_last^T ... A_first^T
// via R <- (diag(g_t) P^T) x R, one wave per chunk, WMMA-chained.
// ---------------------------------------------------------------------------
__global__ void k_chunkprod(const float* __restrict__ y, float* __restrict__ W) {
  const int c = blockIdx.x;                 // 0 .. NCH-2
  const int lane = threadIdx.x;
  const bool hi = lane >= KS;
  const int m = lane & 15;

  float Pc[KS];                             // row m of P^T  (= column m of P)
#pragma unroll
  for (int k = 0; k < KS; ++k) Pc[k] = W[WS_PT + m * KS + k];
  const float mu_m = W[WS_MU + m], is_m = W[WS_IS + m], cf_m = W[WS_CF + m];

  float aLo[4], aHi[4];
#pragma unroll
  for (int j = 0; j < 4; ++j) {
    aLo[j] = hi ? Pc[4 * j + 2] : Pc[4 * j];
    aHi[j] = hi ? Pc[4 * j + 3] : Pc[4 * j + 1];
  }

  v8f R;                                    // running product, D layout; init I
#pragma unroll
  for (int v = 0; v < 8; ++v) R[v] = ((v + (hi ? 8 : 0)) == m) ? 1.0f : 0.0f;

  const int t0 = c * CHUNK + (c == 0 ? 1 : 0);   // chunk 0 starts at t=1
  const int t1 = (c + 1) * CHUNK;
  float yn = y[t0];
  for (int t = t0; t < t1; ++t) {
    const float yc = yn;
    const int tn = (t + 1 < T_LEN) ? t + 1 : T_LEN - 1;
    yn = y[tn];                                       // prefetch next y
    const float z = (yc - mu_m) * is_m;
    const float gm = __expf(-0.5f * z * z) * cf_m;    // g_t[m] for this lane

    Asl A;                                            // A = diag(g_t) P^T
#pragma unroll
    for (int j = 0; j < 4; ++j) { A.a[j].x = gm * aLo[j]; A.a[j].y = gm * aHi[j]; }
    const Bsl B = d_to_b(R, hi);
    const v8f acc = mm_acc(A, B);

    // Cheap wave-uniform rescale: scale only needs to be positive and
    // uniform (it cancels in later normalization).  Use the column-0 sum:
    // partial sum of this lane's 8 rows + SWAP16 pairs lanes l and l^16,
    // so lane 0 then holds sum over all 16 rows of column 0; broadcast it.
    float s = 0.f;
#pragma unroll
    for (int v = 0; v < 8; ++v) s += acc[v];
    s += SWAP16(s);
    const float su = __builtin_bit_cast(
        float, __builtin_amdgcn_readfirstlane(__builtin_bit_cast(int, s)));
    const float sc = __builtin_amdgcn_rcpf(su);
#pragma unroll
    for (int v = 0; v < 8; ++v) R[v] = acc[v] * sc;
  }

  // store N_c^T row-major: element [m][k] at m*16+k
#pragma unroll
  for (int v = 0; v < 8; ++v)
    W[WS_N + c * 256 + (v + (hi ? 8 : 0)) * KS + m] = R[v];
}

// ---------------------------------------------------------------------------
// Phase 2: sequential scan over chunk matrices; one wave, state replicated
// across the 16 columns of V.  Emits normalized start state of each chunk.
// ---------------------------------------------------------------------------
__global__ void k_scan(float* __restrict__ W) {
  const int lane = threadIdx.x;
  const bool hi = lane >= KS;
  const int m = lane & 15;
  const int base = hi ? 8 : 0;

  v8f V;                                    // V[row m][col c] = state[m], all cols equal
#pragma unroll
  for (int v = 0; v < 8; ++v) V[v] = W[WS_STATE + base + v];

  for (int c = 0; c < NCH - 1; ++c) {
    __builtin_prefetch(W + WS_N + (c + 1) * 256 + m * KS, 0, 1);
    float a16[16];
#pragma unroll
    for (int k = 0; k < 16; ++k) a16[k] = W[WS_N + c * 256 + m * KS + k];
    const Asl A = make_a(a16, hi);          // A = N_c^T
    const Bsl B = d_to_b(V, hi);
    const v8f acc = mm_acc(A, B);

    float s = 0.f;
#pragma unroll
    for (int v = 0; v < 8; ++v) s += acc[v];
    s += SWAP16(s);                          // column sum (all cols identical)
    const float inv = 1.0f / s;
#pragma unroll
    for (int v = 0; v < 8; ++v) V[v] = acc[v] * inv;

    if (m == 0) {                            // lanes 0 and 16: store 16-float state
      float* sp = W + WS_STATE + (c + 1) * KS + base;
      v4f lo, hi4;
      lo.x = V[0]; lo.y = V[1]; lo.z = V[2]; lo.w = V[3];
      hi4.x = V[4]; hi4.y = V[5]; hi4.z = V[6]; hi4.w = V[7];
      *(v4f*)sp = lo; *(v4f*)(sp + 4) = hi4;
    }
  }
}

// ---------------------------------------------------------------------------
// Phase 3: emit.  Each wave advances 16 chunks at once: V[state][chunk],
// step: U = P^T x V (WMMA, constant A);  ut = U;  V' = U .* G;  ft = colsum;
// u_norm = V'/ft.  Per-lane predication gates stores only (EXEC full at WMMA).
// ---------------------------------------------------------------------------
__global__ void k_emit(const float* __restrict__ y, float* __restrict__ out,
                       const float* __restrict__ W) {
  const int lane = threadIdx.x;
  const bool hi = lane >= KS;
  const int ci = lane & 15;
  const int base = hi ? 8 : 0;
  const int c = blockIdx.x * 16 + ci;
  const int cc = (c < NCH) ? c : NCH - 1;

  v8f V;
#pragma unroll
  for (int v = 0; v < 8; ++v) V[v] = W[WS_STATE + cc * KS + base + v];

  float muv[8], isv[8], cfv[8];
#pragma unroll
  for (int v = 0; v < 8; ++v) {
    muv[v] = W[WS_MU + base + v];
    isv[v] = W[WS_IS + base + v];
    cfv[v] = W[WS_CF + base + v];
  }

  float Pc[16];                              // row ci of P^T
#pragma unroll
  for (int k = 0; k < 16; ++k) Pc[k] = W[WS_PT + ci * KS + k];
  const Asl A = make_a(Pc, hi);              // constant A = P^T

  const int tstart = c * CHUNK;
  float yn = y[(tstart < T_LEN) ? tstart : T_LEN - 1];
  for (int i = 0; i < CHUNK; ++i) {
    const int t = tstart + i;
    const bool act = (c < NCH) && (t < T_LEN) && !(c == 0 && i == 0);
    const float yc = yn;
    const int tn0 = t + 1;
    yn = y[(tn0 < T_LEN) ? tn0 : T_LEN - 1];

    const Bsl B = d_to_b(V, hi);
    const v8f acc = mm_acc(A, B);            // acc[m][c] = ut_t[m] (predictive)

    if (act) {
      float* po = out + OFF_UT + t * KS + base;
      v4f lo, h4;
      lo.x = acc[0]; lo.y = acc[1]; lo.z = acc[2]; lo.w = acc[3];
      h4.x = acc[4]; h4.y = acc[5]; h4.z = acc[6]; h4.w = acc[7];
      *(v4f*)po = lo; *(v4f*)(po + 4) = h4;
    }

    v8f Vp;
    float s = 0.f;
#pragma unroll
    for (int v = 0; v < 8; ++v) {
      const float z = (yc - muv[v]) * isv[v];
      const float g = __expf(-0.5f * z * z) * cfv[v];
      Vp[v] = acc[v] * g;
      s += Vp[v];
    }
    s += SWAP16(s);                          // full column sum = ft_t
    const float inv = 1.0f / s;
    v8f Vn;
#pragma unroll
    for (int v = 0; v < 8; ++v) Vn[v] = Vp[v] * inv;

    if (act) {
      float* pu = out + OFF_UN + t * KS + base;
      v4f lo, h4;
      lo.x = Vn[0]; lo.y = Vn[1]; lo.z = Vn[2]; lo.w = Vn[3];
      h4.x = Vn[4]; h4.y = Vn[5]; h4.z = Vn[6]; h4.w = Vn[7];
      *(v4f*)pu = lo; *(v4f*)(pu + 4) = h4;
      if (!hi) out[OFF_FT + t] = s;
    }
#pragma unroll
    for (int v = 0; v < 8; ++v) V[v] = act ? Vn[v] : V[v];
  }
}

// ---------------------------------------------------------------------------
extern "C" void kernel_launch(void* const* d_in, const int* in_sizes, int n_in,
                              void* d_out, int out_size, void* d_ws, size_t ws_size,
                              hipStream_t stream) {
  const float* y      = (const float*)d_in[0];
  const float* logits = (const float*)d_in[1];
  const float* mu     = (const float*)d_in[2];
  const float* logsig = (const float*)d_in[3];
  float* out = (float*)d_out;
  float* W   = (float*)d_ws;   // needs (WS_N + NCH*256)*4 ~= 776 KB

  k_init<<<1, 32, 0, stream>>>(y, logits, mu, logsig, out, W);
  k_chunkprod<<<NCH - 1, 32, 0, stream>>>(y, W);
  k_scan<<<1, 32, 0, stream>>>(W);
  k_emit<<<(NCH + 15) / 16, 32, 0, stream>>>(y, out, W);
}